// LSHAttention_8289286881986
// MI455X (gfx1250) — compile-verified
//
#include <hip/hip_runtime.h>

typedef float v2f __attribute__((ext_vector_type(2)));
typedef float v8f __attribute__((ext_vector_type(8)));

#define E_ 512
#define H_ 8
#define D_ 64
#define S_ 2048
#define B_ 4
#define NH_ 5
#define LOG2E_ 1.4426950408889634f
#define INV_SQRT_E_ 0.04419417382415922f  /* 1/sqrt(512) */

static __device__ __forceinline__ v8f wmma_f32(v2f a, v2f b, v8f c) {
  // D = A(16x4) * B(4x16) + C, full fp32
  return __builtin_amdgcn_wmma_f32_16x16x4_f32(false, a, false, b, (short)0, c,
                                               false, false);
}

// ---- CDNA5 async global->LDS (ASYNCcnt-tracked, bypasses VGPRs) -----------
// Generic pointers to __shared__ carry the LDS offset in their low 32 bits
// (ISA 10.2: LDS aperture).
static __device__ __forceinline__ void async_ld_b128(void* lds,
                                                     const void* gsrc) {
  const unsigned ldsa = (unsigned)(size_t)lds;
  asm volatile("global_load_async_to_lds_b128 %0, %1, off" ::"v"(ldsa),
               "v"(gsrc)
               : "memory");
}
static __device__ __forceinline__ void async_ld_b32(void* lds,
                                                    const void* gsrc) {
  const unsigned ldsa = (unsigned)(size_t)lds;
  asm volatile("global_load_async_to_lds_b32 %0, %1, off" ::"v"(ldsa),
               "v"(gsrc)
               : "memory");
}
static __device__ __forceinline__ void wait_async0() {
  asm volatile("s_wait_asynccnt 0x0" ::: "memory");
}

// ---------------------------------------------------------------------------
// Kernel 1: q = x@Wq + bq, v = x@Wv + bv.  One wave per 16x16 output tile.
// A layout (32-bit, 16x4): lanes 0-15 -> M=lane, K = {k, k+1} in vgpr {0,1};
//                          lanes 16-31 -> M=lane-16, K = {k+2, k+3}.
// B layout (4x16): lanes 0-15 -> N=lane, K = {k,k+1}; lanes 16-31 -> K={k+2,k+3}.
// C/D layout: vgpr g, lanes 0-15 -> (M=g, N=lane); lanes 16-31 -> (M=g+8, N=lane-16).
// ---------------------------------------------------------------------------
__global__ __launch_bounds__(128) void lsh_proj_kernel(
    const float* __restrict__ x, const float* __restrict__ Wq,
    const float* __restrict__ bq, const float* __restrict__ Wv,
    const float* __restrict__ bv, float* __restrict__ qbuf,
    float* __restrict__ vbuf) {
  const int lane = threadIdx.x & 31;
  const int wave = threadIdx.x >> 5;
  const int ln   = lane & 15;
  const int half = lane >> 4;
  const int row0 = blockIdx.x * 16;
  const int col0 = blockIdx.y * 64 + wave * 16;
  const bool isV = (blockIdx.z != 0);
  const float* __restrict__ W    = isV ? Wv : Wq;
  const float* __restrict__ bias = isV ? bv : bq;
  float* __restrict__ outp       = isV ? vbuf : qbuf;

  const float* arow = x + (size_t)(row0 + ln) * E_;
  v8f acc = {0.f, 0.f, 0.f, 0.f, 0.f, 0.f, 0.f, 0.f};

#pragma unroll 4
  for (int k = 0; k < E_; k += 4) {
    const int d = k + half * 2;
    const v2f a = *(const v2f*)(arow + d);  // b64, 8B aligned
    v2f b;
    b.x = W[(size_t)d * E_ + col0 + ln];
    b.y = W[(size_t)(d + 1) * E_ + col0 + ln];
    acc = wmma_f32(a, b, acc);
  }

  const float bsc = bias[col0 + ln];
#pragma unroll
  for (int g = 0; g < 8; ++g) {
    outp[(size_t)(row0 + g + half * 8) * E_ + col0 + ln] = acc[g] + bsc;
  }
}

// ---------------------------------------------------------------------------
// Kernel 2: LSH buckets. One thread per (b,s,h).  bkt stored as [B,H,S].
// ---------------------------------------------------------------------------
__global__ __launch_bounds__(256) void lsh_bucket_kernel(
    const float* __restrict__ qbuf, const float* __restrict__ hp,
    int* __restrict__ bkt) {
  const int idx = blockIdx.x * 256 + threadIdx.x;  // (b*S + s)*H + h
  const int h  = idx & (H_ - 1);
  const int bs = idx >> 3;  // b*S + s
  const int s  = bs & (S_ - 1);
  const int b  = bs >> 11;  // / S_

  const float* qp = qbuf + (size_t)bs * E_ + h * D_;
  float proj[NH_];
#pragma unroll
  for (int y = 0; y < NH_; ++y) proj[y] = hp[D_ * NH_ + y];  // ones row
  for (int d = 0; d < D_; ++d) {
    const float qd = qp[d];
#pragma unroll
    for (int y = 0; y < NH_; ++y) proj[y] += qd * hp[d * NH_ + y];
  }
  int bucket = 0;
#pragma unroll
  for (int y = 0; y < NH_; ++y)
    if (proj[y] >= 0.f) bucket |= (1 << y);
  bkt[((size_t)(b * H_ + h)) * S_ + s] = bucket;
}

// ---------------------------------------------------------------------------
// Kernel 3: flash-style LSH attention, 32-key tiles, double-buffered
// async global->LDS staging.  Block = 8 waves = 128 query rows.
// ---------------------------------------------------------------------------
#define JT_ (S_ / 32)  // 64 key-tile steps of 32 keys

__global__ __launch_bounds__(256) void lsh_attn_kernel(
    const float* __restrict__ qbuf, const float* __restrict__ vbuf,
    const int* __restrict__ bkt, float* __restrict__ out) {
  __shared__ float kT[2][32 * 64];  // K tiles, row-major [j][d]
  __shared__ float vT[2][32 * 64];  // V tiles
  __shared__ int   bkS[2][32];      // key buckets
  __shared__ float pS[8][16 * 32];  // per-wave P scratch (row-major 16x32)

  const int b = blockIdx.z, h = blockIdx.y;
  const int q0   = blockIdx.x * 128;
  const int tid  = threadIdx.x;
  const int wave = tid >> 5, lane = tid & 31;
  const int ln   = lane & 15, half = lane >> 4;
  const int qrowBase = q0 + wave * 16;

  const size_t bhS = (size_t)(b * H_ + h) * S_;

  // Resident A-operand: this wave's 16 Q rows in A layout (32 VGPRs)
  const float* qrow = qbuf + ((size_t)(b * S_ + qrowBase + ln)) * E_ + h * D_;
  v2f aq[16];
#pragma unroll
  for (int kk = 0; kk < 16; ++kk)
    aq[kk] = *(const v2f*)(qrow + kk * 4 + half * 2);

  // Per-lane query-row buckets (rows g+half*8 for g=0..7)
  int bq_[8];
  const int* bb = bkt + bhS + qrowBase + half * 8;
#pragma unroll
  for (int g = 0; g < 8; ++g) bq_[g] = bb[g];

  float m_[8], l_[8];
  v8f O[4];
  const v8f vzero = {0.f, 0.f, 0.f, 0.f, 0.f, 0.f, 0.f, 0.f};
#pragma unroll
  for (int g = 0; g < 8; ++g) { m_[g] = -3.0e38f; l_[g] = 0.f; }
#pragma unroll
  for (int ct = 0; ct < 4; ++ct) O[ct] = vzero;

  float* ps = &pS[wave][0];

  // --- async stage of one 32-key tile (8KB K + 8KB V + 32 buckets) ---------
  auto stage = [&](int jt, int buf) {
    const int j0 = jt * 32;
#pragma unroll
    for (int i = 0; i < 2; ++i) {
      const int idx = tid + i * 256;      // 0..511
      const int r   = idx >> 4;           // key row 0..31
      const int c4  = (idx & 15) * 4;     // 0..60
      const size_t src = ((size_t)(b * S_ + j0 + r)) * E_ + h * D_ + c4;
      async_ld_b128(&kT[buf][r * 64 + c4], qbuf + src);
      async_ld_b128(&vT[buf][r * 64 + c4], vbuf + src);
    }
    if (tid < 32) async_ld_b32(&bkS[buf][tid], bkt + bhS + j0 + tid);
  };

  stage(0, 0);
  wait_async0();
  __syncthreads();

  for (int jt = 0; jt < JT_; ++jt) {
    const int buf = jt & 1;
    if (jt + 1 < JT_) stage(jt + 1, buf ^ 1);  // prefetch next tile

    const float* kTb = &kT[buf][0];
    const float* vTb = &vT[buf][0];

    // ---- S = Q @ K^T, two 16x16 score tiles (32 WMMA) ----
    v8f c0 = vzero, c1 = vzero;
#pragma unroll
    for (int kk = 0; kk < 16; ++kk) {
      const int d = kk * 4 + half * 2;
      const v2f b0 = *(const v2f*)(kTb + ln * 64 + d);
      const v2f b1 = *(const v2f*)(kTb + (16 + ln) * 64 + d);
      c0 = wmma_f32(aq[kk], b0, c0);
      c1 = wmma_f32(aq[kk], b1, c1);
    }

    // ---- logits with LSH multiplier ----
    const int bkn0 = bkS[buf][ln];
    const int bkn1 = bkS[buf][16 + ln];
    float p0[8], p1[8], tmax[8];
#pragma unroll
    for (int g = 0; g < 8; ++g) {
      const float f0 = (bq_[g] == bkn0) ? 31.0f : 30.0f;  // (C-2)+same
      const float f1 = (bq_[g] == bkn1) ? 31.0f : 30.0f;
      p0[g] = c0[g] * (INV_SQRT_E_ * f0);
      p1[g] = c1[g] * (INV_SQRT_E_ * f1);
    }

    // ---- row max over 32 keys (16-lane-half butterflies) ----
#pragma unroll
    for (int g = 0; g < 8; ++g) {
      float xv = fmaxf(p0[g], p1[g]);
      xv = fmaxf(xv, __shfl_xor(xv, 1, 32));
      xv = fmaxf(xv, __shfl_xor(xv, 2, 32));
      xv = fmaxf(xv, __shfl_xor(xv, 4, 32));
      xv = fmaxf(xv, __shfl_xor(xv, 8, 32));
      tmax[g] = xv;
    }

    float alpha[8];
#pragma unroll
    for (int g = 0; g < 8; ++g) {
      const float mn = fmaxf(m_[g], tmax[g]);
      alpha[g] = __builtin_amdgcn_exp2f((m_[g] - mn) * LOG2E_);
      m_[g] = mn;
    }
#pragma unroll
    for (int g = 0; g < 8; ++g) {
      p0[g] = __builtin_amdgcn_exp2f((p0[g] - m_[g]) * LOG2E_);
      p1[g] = __builtin_amdgcn_exp2f((p1[g] - m_[g]) * LOG2E_);
    }

    // ---- row sums + running denominator ----
#pragma unroll
    for (int g = 0; g < 8; ++g) {
      float sv = p0[g] + p1[g];
      sv += __shfl_xor(sv, 1, 32);
      sv += __shfl_xor(sv, 2, 32);
      sv += __shfl_xor(sv, 4, 32);
      sv += __shfl_xor(sv, 8, 32);
      l_[g] = l_[g] * alpha[g] + sv;
    }

    // ---- rescale accumulators (per-row, matches C/D row mapping) ----
#pragma unroll
    for (int ct = 0; ct < 4; ++ct)
#pragma unroll
      for (int g = 0; g < 8; ++g) O[ct][g] *= alpha[g];

    // ---- P: C/D layout -> LDS row-major 16x32 -> reload in A layout ----
#pragma unroll
    for (int g = 0; g < 8; ++g) {
      ps[(g + half * 8) * 32 + ln]      = p0[g];
      ps[(g + half * 8) * 32 + 16 + ln] = p1[g];
    }
    // DS ops are in-order within a wave: loads below see the stores above.
#pragma unroll
    for (int kk = 0; kk < 8; ++kk) {
      const int kx = kk * 4 + half * 2;
      const v2f ap = *(const v2f*)(ps + ln * 32 + kx);
#pragma unroll
      for (int ct = 0; ct < 4; ++ct) {
        v2f bv2;
        bv2.x = vTb[kx * 64 + ct * 16 + ln];
        bv2.y = vTb[(kx + 1) * 64 + ct * 16 + ln];
        O[ct] = wmma_f32(ap, bv2, O[ct]);
      }
    }

    wait_async0();    // this wave's prefetch for jt+1 landed
    __syncthreads();  // everyone done with buf, next tile fully staged
  }

  // ---- finalize: divide by softmax denominator, write [B,S,H*D] ----
#pragma unroll
  for (int g = 0; g < 8; ++g) l_[g] = 1.0f / l_[g];
#pragma unroll
  for (int ct = 0; ct < 4; ++ct) {
#pragma unroll
    for (int g = 0; g < 8; ++g) {
      out[((size_t)(b * S_ + qrowBase + g + half * 8)) * E_ + h * D_ +
          ct * 16 + ln] = O[ct][g] * l_[g];
    }
  }
}

// ---------------------------------------------------------------------------
extern "C" void kernel_launch(void* const* d_in, const int* in_sizes, int n_in,
                              void* d_out, int out_size, void* d_ws,
                              size_t ws_size, hipStream_t stream) {
  const float* x  = (const float*)d_in[0];
  const float* Wq = (const float*)d_in[1];
  const float* bq = (const float*)d_in[2];
  const float* Wv = (const float*)d_in[3];
  const float* bv = (const float*)d_in[4];
  const float* hp = (const float*)d_in[5];
  float* out = (float*)d_out;

  // Workspace: q (16MB) | v (16MB) | buckets (256KB)
  float* qbuf = (float*)d_ws;
  float* vbuf = qbuf + (size_t)B_ * S_ * E_;
  int*   bkt  = (int*)(vbuf + (size_t)B_ * S_ * E_);

  lsh_proj_kernel<<<dim3(B_ * S_ / 16, E_ / 64, 2), 128, 0, stream>>>(
      x, Wq, bq, Wv, bv, qbuf, vbuf);
  lsh_bucket_kernel<<<dim3(B_ * S_ * H_ / 256), 256, 0, stream>>>(qbuf, hp, bkt);
  lsh_attn_kernel<<<dim3(S_ / 128, H_, B_), 256, 0, stream>>>(qbuf, vbuf, bkt,
                                                              out);
}